// CompositeEmbeddingA_26749056320252
// MI455X (gfx1250) — compile-verified
//
#include <hip/hip_runtime.h>

typedef float v2f __attribute__((ext_vector_type(2)));
typedef float v8f __attribute__((ext_vector_type(8)));

#define S_TOTAL 37448   // sum(LAYER_SIZES)
#define T_OUT   5704    // 8 + 64 + 512 + 1024 + 4096
#define EMB     256

// ---------------------------------------------------------------------------
// Layers 0-2: out[b, s] = val_emb[v] + pos0[p0] + pos1[p1] + pos2[p2] + dep[d]
// One block per token (4672 tokens), 256 threads = one embedding dim each.
// ---------------------------------------------------------------------------
__global__ __launch_bounds__(256)
void embed_small_kernel(
    const int* __restrict__ value, const int* __restrict__ depth,
    const int* __restrict__ position,
    const float* __restrict__ ve0, const float* __restrict__ ve1, const float* __restrict__ ve2,
    const float* __restrict__ pe0, const float* __restrict__ pe1, const float* __restrict__ pe2,
    const float* __restrict__ de0, const float* __restrict__ de1, const float* __restrict__ de2,
    float* __restrict__ out)
{
    const int tt = blockIdx.x;            // 0..4671
    const int b  = tt / 584;
    const int s  = tt - b * 584;
    const float *ve, *pe, *de;
    if (s < 8)       { ve = ve0; pe = pe0; de = de0; }
    else if (s < 72) { ve = ve1; pe = pe1; de = de1; }
    else             { ve = ve2; pe = pe2; de = de2; }
    const int g  = b * S_TOTAL + s;
    const int e  = threadIdx.x;
    const int v  = value[g];
    const int d  = depth[g];
    const int p0 = position[g*3+0], p1 = position[g*3+1], p2 = position[g*3+2];
    const float x = ve[v*EMB + e]
                  + pe[(      p0)*EMB + e]
                  + pe[( 64 + p1)*EMB + e]
                  + pe[(128 + p2)*EMB + e]
                  + de[d*EMB + e];
    out[(b * T_OUT + s) * EMB + e] = x;
}

// ---------------------------------------------------------------------------
// One-shot weight permute into workspace:
//   w2[kidx*256 + o] = conv_w[(o*256 + e)*KSZ + kk],  kidx = kk*256 + e
// Makes the GEMM B-operand row-contiguous so tiles can be streamed to LDS
// with GLOBAL_LOAD_ASYNC_TO_LDS_B128. Total 3 MB -> lives in L2.
// ---------------------------------------------------------------------------
__global__ __launch_bounds__(256)
void permute_w_kernel(const float* __restrict__ convw, float* __restrict__ w2,
                      int KSZ, int KDIM)
{
    const int i = blockIdx.x * 256 + threadIdx.x;    // over KDIM*256
    if (i >= KDIM * 256) return;
    const int kidx = i >> 8;
    const int o    = i & 255;
    const int kk   = kidx >> 8;
    const int e    = kidx & 255;
    w2[i] = convw[(o * 256 + e) * KSZ + kk];
}

// ---------------------------------------------------------------------------
// Layers 3-4: fused embedding-gather + GEMM via V_WMMA_F32_16X16X4_F32.
// Block tile: 64 (M) x 128 (N), 8 waves in a 4x2 grid, each wave 16x64.
// Per 16-wide K chunk (fixed conv tap kk since 256 % 16 == 0):
//   - B chunk (16x128 of permuted weights) streamed via async global->LDS B128
//   - A chunk (64x16) gathered+summed from embedding tables (overlaps asyncs)
//   - s_wait_asynccnt 0 + barrier, then 4 k-steps x 4 n-tiles of fp32 WMMA.
// Token indices for the tile are a contiguous run; raw value/position slices
// are also async-copied to LDS in the prologue (all runs 16B-aligned).
// ---------------------------------------------------------------------------
template<int KSZ, int TPRIME, int IN_START, int OUT_START>
__global__ __launch_bounds__(256)
void conv_wmma_kernel(
    const int*   __restrict__ value,
    const int*   __restrict__ position,
    const float* __restrict__ vemb,   // (4, 256)
    const float* __restrict__ pemb,   // (3, 64, 256)
    const float* __restrict__ w2,     // (KSZ*256, 256) permuted weights
    const float* __restrict__ convb,  // (256,)
    float* __restrict__ out)
{
    constexpr int NCHUNK = KSZ * 16;             // K/16 chunks
    constexpr int NTOK   = 64 * KSZ;             // tokens per tile (256 / 512)
    __shared__ __align__(16) int   vraw[512];    // raw value[] slice
    __shared__ __align__(16) int   praw[1536];   // raw position[] slice
    __shared__ __align__(16) float A_lds[64 * 17];
    __shared__ __align__(16) float B_lds[16 * 144];

    const int tid  = threadIdx.x;
    const int lane = tid & 31;
    const int wave = tid >> 5;
    const int mw   = wave & 3;                   // wave M position (0..3)
    const int nw   = wave >> 2;                  // wave N position (0..1)
    const int half = lane >> 4;                  // lane half (K +0 / +2 select)
    const int lr   = lane & 15;

    const int m0    = blockIdx.x * 64;           // M tile base (64 | TPRIME)
    const int obase = blockIdx.y * 128;          // N tile base

    const int bb   = m0 / TPRIME;
    const int tp0  = m0 - bb * TPRIME;
    const int tok0 = bb * S_TOTAL + IN_START + tp0 * KSZ;   // multiple of 8

    // ---- prologue: async-copy contiguous index runs into LDS ----
    {
        constexpr int NV4 = NTOK / 4;            // value b128 units
        constexpr int NT4 = NV4 + (3 * NTOK) / 4;// total units (256 or 512)
        #pragma unroll
        for (int i = 0; i < (NT4 + 255) / 256; ++i) {
            const int u = tid + i * 256;
            unsigned lds;
            unsigned long long ga;
            if (u < NV4) {
                lds = (unsigned)(uintptr_t)&vraw[u * 4];
                ga  = (unsigned long long)(uintptr_t)(value + tok0 + u * 4);
            } else {
                const int u2 = u - NV4;
                lds = (unsigned)(uintptr_t)&praw[u2 * 4];
                ga  = (unsigned long long)(uintptr_t)(position + tok0 * 3 + u2 * 4);
            }
            asm volatile("global_load_async_to_lds_b128 %0, %1, off"
                         :: "v"(lds), "v"(ga) : "memory");
        }
        asm volatile("s_wait_asynccnt 0x0" ::: "memory");
    }
    __syncthreads();

    v8f acc[4];
    #pragma unroll
    for (int nt = 0; nt < 4; ++nt) acc[nt] = (v8f){0,0,0,0,0,0,0,0};

    const int arow = tid >> 2;                   // A staging: row 0..63
    const int aj   = tid & 3;                    //            4 floats each

    for (int c = 0; c < NCHUNK; ++c) {
        const int kk = c >> 4;                   // conv tap for this chunk
        const int e0 = (c & 15) << 4;            // embedding-dim base

        // ---- async-stage B chunk: 16 rows x 128 cols, 512 b128 units ----
        #pragma unroll
        for (int i = 0; i < 2; ++i) {
            const int u   = tid + i * 256;
            const int row = u >> 5;              // 0..15
            const int c4  = (u & 31) * 4;        // 0..124
            unsigned lds = (unsigned)(uintptr_t)&B_lds[row * 144 + c4];
            unsigned long long ga = (unsigned long long)(uintptr_t)
                (w2 + (c * 16 + row) * 256 + obase + c4);
            asm volatile("global_load_async_to_lds_b128 %0, %1, off"
                         :: "v"(lds), "v"(ga) : "memory");
        }

        // ---- stage A (overlaps the async B copy): gather + sum 4 tables ----
        {
            const int tl = arow * KSZ + kk;
            const int ov = (      vraw[tl]      ) * EMB;
            const int o0 = (      praw[tl*3+0]  ) * EMB;
            const int o1 = ( 64 + praw[tl*3+1]  ) * EMB;
            const int o2 = (128 + praw[tl*3+2]  ) * EMB;
            const int e  = e0 + aj * 4;
            const float4 x0 = *(const float4*)(vemb + ov + e);
            const float4 x1 = *(const float4*)(pemb + o0 + e);
            const float4 x2 = *(const float4*)(pemb + o1 + e);
            const float4 x3 = *(const float4*)(pemb + o2 + e);
            float* dst = &A_lds[arow * 17 + aj * 4];
            dst[0] = x0.x + x1.x + x2.x + x3.x;
            dst[1] = x0.y + x1.y + x2.y + x3.y;
            dst[2] = x0.z + x1.z + x2.z + x3.z;
            dst[3] = x0.w + x1.w + x2.w + x3.w;
        }

        asm volatile("s_wait_asynccnt 0x0" ::: "memory");
        __syncthreads();

        // ---- compute: 4 K-steps of 4, 4 N-tiles each ----
        #pragma unroll
        for (int s = 0; s < 4; ++s) {
            const int kA = s * 4 + 2 * half;     // frag layout: half picks K+2
            v2f a;
            a.x = A_lds[(mw * 16 + lr) * 17 + kA + 0];
            a.y = A_lds[(mw * 16 + lr) * 17 + kA + 1];
            #pragma unroll
            for (int nt = 0; nt < 4; ++nt) {
                const int col = nw * 64 + nt * 16 + lr;
                v2f bf;
                bf.x = B_lds[(kA + 0) * 144 + col];
                bf.y = B_lds[(kA + 1) * 144 + col];
                acc[nt] = __builtin_amdgcn_wmma_f32_16x16x4_f32(
                    false, a, false, bf, (short)0, acc[nt], false, false);
            }
        }
        __syncthreads();                         // done reading LDS this chunk
    }

    // ---- store C tiles (+bias): VGPR r -> (M = r + 8*half, N = lr) ----
    const int mrow_base = m0 + mw * 16;
    #pragma unroll
    for (int nt = 0; nt < 4; ++nt) {
        const int o    = obase + nw * 64 + nt * 16 + lr;
        const float bv = convb[o];
        #pragma unroll
        for (int r = 0; r < 8; ++r) {
            const int m    = mrow_base + r + 8 * half;
            const int bidx = m / TPRIME;
            const int tp   = m - bidx * TPRIME;
            out[(bidx * T_OUT + OUT_START + tp) * EMB + o] = acc[nt][r] + bv;
        }
    }
}

extern "C" void kernel_launch(void* const* d_in, const int* in_sizes, int n_in,
                              void* d_out, int out_size, void* d_ws, size_t ws_size,
                              hipStream_t stream) {
    const int*   value    = (const int*)d_in[0];
    const int*   depth    = (const int*)d_in[1];
    const int*   position = (const int*)d_in[2];
    const float* ve0 = (const float*)d_in[3];
    const float* pe0 = (const float*)d_in[4];
    const float* ve1 = (const float*)d_in[5];
    const float* pe1 = (const float*)d_in[6];
    const float* ve2 = (const float*)d_in[7];
    const float* pe2 = (const float*)d_in[8];
    const float* ve3 = (const float*)d_in[9];
    const float* pe3 = (const float*)d_in[10];
    const float* ve4 = (const float*)d_in[11];
    const float* pe4 = (const float*)d_in[12];
    const float* de0 = (const float*)d_in[13];
    const float* de1 = (const float*)d_in[14];
    const float* de2 = (const float*)d_in[15];
    const float* cw3 = (const float*)d_in[16];
    const float* cb3 = (const float*)d_in[17];
    const float* cw4 = (const float*)d_in[18];
    const float* cb4 = (const float*)d_in[19];
    float* out = (float*)d_out;

    // Workspace: permuted weights, 1 MB (layer 3) + 2 MB (layer 4)
    float* w2a = (float*)d_ws;
    float* w2b = w2a + 1024 * 256;
    permute_w_kernel<<<1024, 256, 0, stream>>>(cw3, w2a, 4, 1024);
    permute_w_kernel<<<2048, 256, 0, stream>>>(cw4, w2b, 8, 2048);

    // Layers 0-2: 8*(8+64+512) = 4672 tokens
    embed_small_kernel<<<4672, 256, 0, stream>>>(
        value, depth, position, ve0, ve1, ve2, pe0, pe1, pe2, de0, de1, de2, out);

    // Layer 3: M = 8192 -> 128 M-tiles, 2 N-tiles
    conv_wmma_kernel<4, 1024,  584,  584><<<dim3(128, 2), 256, 0, stream>>>(
        value, position, ve3, pe3, w2a, cb3, out);

    // Layer 4: M = 32768 -> 512 M-tiles, 2 N-tiles
    conv_wmma_kernel<8, 4096, 4680, 1608><<<dim3(512, 2), 256, 0, stream>>>(
        value, position, ve4, pe4, w2b, cb4, out);
}